// SlaterPooling_13735305412839
// MI455X (gfx1250) — compile-verified
//
#include <hip/hip_runtime.h>

// SlaterPooling on MI455X (gfx1250, wave32):
//   out[b][c] = det(Xup[b][:, cfg_up[c]]) * det(Xdn[b][:, cfg_dn[c]])
// B=8192, nspin=16, NMO=64, NCONF=64.
//
// Strategy:
//  - gather-as-matmul: sub = X(16x64) * P(64x16 one-hot), exact in fp32,
//    done with 16 chained V_WMMA_F32_16X16X4_F32 per spin. X rows are loaded
//    contiguously in WMMA A-operand layout ONCE per batch and reused across
//    all 64 configs from registers. One-hot B operands are synthesized with
//    compares (no memory).
//  - config tables (8 KB) staged to LDS once per block; the config loop's
//    only VMEM op is the final store.
//  - both 16x16 determinants of a (b,c) pair are computed by ONE wave:
//    lanes 0-15 hold up-matrix columns, lanes 16-31 down-matrix columns
//    (redistributed from WMMA C-layout with ds_swizzle SWAPX16).
//    Unrolled LU: one ds_bpermute broadcast per multiplier, one FMA updates
//    a whole row; pivot reciprocal = v_rcp_f32 + 1 Newton step;
//    det = product of pivots; final up*down via SWAPX16.

#define B_TOT   8192
#define NMO     64
#define NCONF   64
#define NSPIN   16
#define WAVES_PER_BLOCK 8
#define BLOCK_THREADS   (32 * WAVES_PER_BLOCK)
#define CFG_INTS        (NCONF * NSPIN)          // 1024 per spin

typedef float v8f __attribute__((ext_vector_type(8)));
typedef float v2f __attribute__((ext_vector_type(2)));

__device__ __forceinline__ float swap16(float x) {
  // ds_swizzle SWAPX16: xor=0x10, or=0, and=0x1f  -> offset 0x401f
  return __int_as_float(__builtin_amdgcn_ds_swizzle(__float_as_int(x), 0x401f));
}

__device__ __forceinline__ float bcast_half(float x, int k, int hibit) {
  // broadcast from lane k (lower half) / lane 16+k (upper half)
  int idx = (hibit | k) << 2;   // byte index for ds_bpermute
  return __int_as_float(__builtin_amdgcn_ds_bpermute(idx, __float_as_int(x)));
}

__device__ __forceinline__ float fast_rcp(float p) {
  // v_rcp_f32 + one Newton-Raphson refinement (~1 ulp)
  float r = __builtin_amdgcn_rcpf(p);
  return fmaf(fmaf(-p, r, 1.0f), r, r);
}

__global__ __launch_bounds__(BLOCK_THREADS, 1) void slater_det_kernel(
    const float* __restrict__ x,        // (B, 32, 64) f32
    const int*   __restrict__ cfg_up,   // (64, 16)
    const int*   __restrict__ cfg_dn,   // (64, 16)
    float*       __restrict__ out) {    // (B, 64)
  __shared__ int scfg[2 * CFG_INTS];    // [0,1024): up, [1024,2048): down

  const int tid  = threadIdx.x;
  const int lane = tid & 31;
  const int wave = tid >> 5;
  const int b    = blockIdx.x * WAVES_PER_BLOCK + wave;

  const int col = lane & 15;        // column / row index this lane serves
  const int hi  = lane & 16;        // 0 = lower half-wave, 16 = upper
  const int kb  = (lane >> 4) << 1; // A/B operand K sub-offset: 0 or 2

  // ---- stage config tables into LDS (once per block) ----
#pragma unroll
  for (int i = tid; i < 2 * CFG_INTS; i += BLOCK_THREADS)
    scfg[i] = (i < CFG_INTS) ? cfg_up[i] : cfg_dn[i - CFG_INTS];
  __syncthreads();

  // ---- X rows in WMMA A-operand layout, loaded once per batch ----
  // A 16x4 f32 layout: lanes 0-15 hold row M=lane with K=0,1 in vgpr0,1;
  // lanes 16-31 hold the same rows with K=2,3.
  v2f xu[16], xd[16];
  const float* xrow_u = x + ((size_t)b * 32 + col) * NMO;
  const float* xrow_d = x + ((size_t)b * 32 + NSPIN + col) * NMO;
#pragma unroll
  for (int t = 0; t < 16; ++t) {
    xu[t] = *reinterpret_cast<const v2f*>(xrow_u + 4 * t + kb);
    xd[t] = *reinterpret_cast<const v2f*>(xrow_d + 4 * t + kb);
  }

  for (int c = 0; c < NCONF; ++c) {
    // orbital index selected for this lane's output column (from LDS)
    const int cju = scfg[c * NSPIN + col];
    const int cjd = scfg[CFG_INTS + c * NSPIN + col];

    v8f cu = {0.f, 0.f, 0.f, 0.f, 0.f, 0.f, 0.f, 0.f};
    v8f cd = {0.f, 0.f, 0.f, 0.f, 0.f, 0.f, 0.f, 0.f};

    // ---- sub = X * P  via 16 chained fp32 WMMAs (K=4 each), per spin ----
    // B 4x16 layout: vgpr0 = rows K=4t+0 (lanes 0-15) / 4t+2 (lanes 16-31),
    //                vgpr1 = rows K=4t+1 / 4t+3.  One-hot: P[k][j]=(cfg[j]==k)
#pragma unroll
    for (int t = 0; t < 16; ++t) {
      const int k0 = 4 * t + kb;
      v2f bu, bd;
      bu.x = (cju == k0)     ? 1.0f : 0.0f;
      bu.y = (cju == k0 + 1) ? 1.0f : 0.0f;
      bd.x = (cjd == k0)     ? 1.0f : 0.0f;
      bd.y = (cjd == k0 + 1) ? 1.0f : 0.0f;
      cu = __builtin_amdgcn_wmma_f32_16x16x4_f32(
          false, xu[t], false, bu, (short)0, cu, false, false);
      cd = __builtin_amdgcn_wmma_f32_16x16x4_f32(
          false, xd[t], false, bd, (short)0, cd, false, false);
    }

    // ---- C-layout -> column-per-lane (up in lanes 0-15, down in 16-31) ----
    // C layout: vgpr v = row v (lanes 0-15) / row v+8 (lanes 16-31), col=lane%16
    float a[16];
#pragma unroll
    for (int i = 0; i < 8; ++i) {
      const float td = swap16(cd[i]);     // down rows 0-7 -> upper half
      a[i] = (hi == 0) ? cu[i] : td;
      const float tu = swap16(cu[i]);     // up rows 8-15 -> lower half
      a[i + 8] = (hi == 0) ? tu : cd[i];
    }

    // ---- LU (no pivoting), both halves in lockstep ----
    float det = 1.0f;
#pragma unroll
    for (int k = 0; k < 16; ++k) {
      const float p = bcast_half(a[k], k, hi);   // pivot a[k][k]
      det *= p;
      if (k < 15) {
        const float u = a[k] * fast_rcp(p);      // scaled pivot row
#pragma unroll
        for (int i = k + 1; i < 16; ++i) {
          const float m = bcast_half(a[i], k, hi);  // a[i][k]
          a[i] = fmaf(-m, u, a[i]);                 // trailing update, all cols
        }
      }
    }

    const float res = det * swap16(det);  // det_up * det_down (both halves)
    if (lane == 0) out[(size_t)b * NCONF + c] = res;
  }
}

extern "C" void kernel_launch(void* const* d_in, const int* in_sizes, int n_in,
                              void* d_out, int out_size, void* d_ws, size_t ws_size,
                              hipStream_t stream) {
  (void)in_sizes; (void)n_in; (void)d_ws; (void)ws_size; (void)out_size;
  const float* x   = (const float*)d_in[0];
  const int*   cu  = (const int*)d_in[1];
  const int*   cdn = (const int*)d_in[2];
  float*       out = (float*)d_out;

  dim3 grid(B_TOT / WAVES_PER_BLOCK);
  dim3 block(BLOCK_THREADS);
  hipLaunchKernelGGL(slater_det_kernel, grid, block, 0, stream, x, cu, cdn, out);
}